// FrontDoorBackboneLayer_38508676776169
// MI455X (gfx1250) — compile-verified
//
#include <hip/hip_runtime.h>
#include <hip/hip_bf16.h>

// GAT layer for MI455X (gfx1250), wave32.
//   h = x @ W                      (WMMA f32 16x16x4, W^T staged in LDS)
//   a_src = h @ att[:D], a_dst = h @ att[D:]   (fused into GEMM epilogue)
//   logits = leaky_relu(a_src[src] + a_dst[dst]); softmax-normalized aggregate
//   out = segsum(e * h[dst], src) / (segsum(e, src) + eps) + x

typedef __attribute__((ext_vector_type(2))) float f32x2;
typedef __attribute__((ext_vector_type(4))) float f32x4;
typedef __attribute__((ext_vector_type(8))) float f32x8;

#define GAT_D 128
#define LEAKY 0.01f
#define GAT_EPS 1e-8f

__device__ __forceinline__ void atomic_add_f32(float* p, float v) {
  // relaxed, agent scope -> global_atomic_add_f32 (no CAS loop)
  __hip_atomic_fetch_add(p, v, __ATOMIC_RELAXED, __HIP_MEMORY_SCOPE_AGENT);
}

__device__ __forceinline__ unsigned f32_mono(float f) {
  unsigned u = __float_as_uint(f);
  return (u & 0x80000000u) ? ~u : (u | 0x80000000u);
}
__device__ __forceinline__ float f32_unmono(unsigned u) {
  return (u & 0x80000000u) ? __uint_as_float(u & 0x7FFFFFFFu)
                           : __uint_as_float(~u);
}

// ---------------------------------------------------------------------------
// Kernel 1: h = x @ W via V_WMMA_F32_16X16X4_F32; fused a_src/a_dst epilogue.
// Block = 256 threads = 8 waves; each wave computes a 16-row x 128-col strip.
// ---------------------------------------------------------------------------
__global__ __launch_bounds__(256) void gat_gemm_wmma_f32(
    const float* __restrict__ x, const float* __restrict__ w,
    const float* __restrict__ att, float* __restrict__ h,
    float* __restrict__ a_src, float* __restrict__ a_dst, int n)
{
  // W transposed in LDS: wT[col][k], stride 130 floats (even -> 8B aligned v2f)
  __shared__ float wT[GAT_D * 130];

  const int tid  = threadIdx.x;
  const int wave = tid >> 5;
  const int lane = tid & 31;
  const int half = lane >> 4;   // 0: lanes 0-15, 1: lanes 16-31
  const int l16  = lane & 15;

  // Cooperative transpose-load of W (128x128 f32) into LDS.
  for (int idx = tid; idx < GAT_D * (GAT_D / 4); idx += 256) {
    const int k  = idx >> 5;          // row of W
    const int n4 = (idx & 31) * 4;    // col group
    f32x4 wv = *(const f32x4*)(w + k * GAT_D + n4);
    wT[(n4 + 0) * 130 + k] = wv.x;
    wT[(n4 + 1) * 130 + k] = wv.y;
    wT[(n4 + 2) * 130 + k] = wv.z;
    wT[(n4 + 3) * 130 + k] = wv.w;
  }
  __syncthreads();

  const int rowBase = blockIdx.x * 128 + wave * 16;

  // A-matrix row for this lane (clamped; stores are guarded below).
  int arow = rowBase + l16;
  if (arow >= n) arow = n - 1;
  const float* xrow = x + (size_t)arow * GAT_D;

  f32x8 acc[8] = {};

  for (int k0 = 0; k0 < GAT_D; k0 += 4) {
    const int ksel = k0 + half * 2;
    // A 16x4 frag: VGPR0={K0,K2}, VGPR1={K1,K3} per half-wave.
    f32x2 a = *(const f32x2*)(xrow + ksel);
    #pragma unroll
    for (int t = 0; t < 8; ++t) {
      const int col = t * 16 + l16;
      // B 4x16 frag, same K striping; contiguous in transposed LDS.
      f32x2 b = *(const f32x2*)(&wT[col * 130 + ksel]);
      acc[t] = __builtin_amdgcn_wmma_f32_16x16x4_f32(
          /*neg_a=*/false, a, /*neg_b=*/false, b,
          /*c_mod=*/(short)0, acc[t], /*reuse_a=*/false, /*reuse_b=*/false);
    }
  }

  if (rowBase < n) {
    // Store h: C/D layout -> row = i + 8*half, col = t*16 + l16.
    #pragma unroll
    for (int t = 0; t < 8; ++t) {
      const int c = t * 16 + l16;
      #pragma unroll
      for (int i = 0; i < 8; ++i) {
        const int r = rowBase + i + half * 8;
        if (r < n) h[(size_t)r * GAT_D + c] = acc[t][i];
      }
    }
    // Fused attention GEMVs: a_src[r] = sum_c h[r][c]*att[c], a_dst with att[D+c].
    #pragma unroll
    for (int i = 0; i < 8; ++i) {
      float ps = 0.f, pd = 0.f;
      #pragma unroll
      for (int t = 0; t < 8; ++t) {
        const int c = t * 16 + l16;
        ps += acc[t][i] * att[c];
        pd += acc[t][i] * att[GAT_D + c];
      }
      // Reduce across the 16 lanes of this half-wave.
      #pragma unroll
      for (int off = 1; off < 16; off <<= 1) {
        ps += __shfl_xor(ps, off, 32);
        pd += __shfl_xor(pd, off, 32);
      }
      if (l16 == 0) {
        const int r = rowBase + i + half * 8;
        if (r < n) { a_src[r] = ps; a_dst[r] = pd; }
      }
    }
  }
}

// ---------------------------------------------------------------------------
// Kernel 2: global max of leaky_relu(a_src[src]+a_dst[dst]) over E+N edges.
// ---------------------------------------------------------------------------
__global__ __launch_bounds__(256) void gat_logit_max(
    const int* __restrict__ src, const int* __restrict__ dst,
    const float* __restrict__ a_src, const float* __restrict__ a_dst,
    unsigned* __restrict__ gmax, int E, int n)
{
  const int total = E + n;
  float m = -3.4028235e38f;
  for (int i = blockIdx.x * blockDim.x + threadIdx.x; i < total;
       i += gridDim.x * blockDim.x) {
    const int s = (i < E) ? src[i] : (i - E);
    const int d = (i < E) ? dst[i] : (i - E);
    const float z = a_src[s] + a_dst[d];
    const float lg = (z > 0.f) ? z : LEAKY * z;
    m = fmaxf(m, lg);
  }
  #pragma unroll
  for (int off = 16; off > 0; off >>= 1) m = fmaxf(m, __shfl_xor(m, off, 32));
  if ((threadIdx.x & 31) == 0) atomicMax(gmax, f32_mono(m));
}

// ---------------------------------------------------------------------------
// Kernel 3: one wave per edge. e = exp(lrelu(a_src[s]+a_dst[d]) - max);
// denom[s] += e; agg[s][:] += e * h[d][:]  (L2-resident f32 atomics).
// ---------------------------------------------------------------------------
__global__ __launch_bounds__(256) void gat_edge_aggregate(
    const int* __restrict__ src, const int* __restrict__ dst,
    const float* __restrict__ a_src, const float* __restrict__ a_dst,
    const float* __restrict__ h, const unsigned* __restrict__ gmax,
    float* __restrict__ agg, float* __restrict__ denom, int E, int n)
{
  const int wid  = blockIdx.x * (blockDim.x >> 5) + (threadIdx.x >> 5);
  const int lane = threadIdx.x & 31;
  const int total = E + n;
  if (wid >= total) return;

  const int s = (wid < E) ? src[wid] : (wid - E);
  const int d = (wid < E) ? dst[wid] : (wid - E);

  const float m  = f32_unmono(*gmax);
  const float z  = a_src[s] + a_dst[d];
  const float lg = (z > 0.f) ? z : LEAKY * z;
  const float e  = __expf(lg - m);

  if (lane == 0) atomic_add_f32(&denom[s], e);

  const float* hd = h + (size_t)d * GAT_D;
  float* os = agg + (size_t)s * GAT_D;
  f32x4 hv = *(const f32x4*)(hd + lane * 4);
  atomic_add_f32(os + lane * 4 + 0, e * hv.x);
  atomic_add_f32(os + lane * 4 + 1, e * hv.y);
  atomic_add_f32(os + lane * 4 + 2, e * hv.z);
  atomic_add_f32(os + lane * 4 + 3, e * hv.w);
}

// ---------------------------------------------------------------------------
// Kernel 4: out = agg / (denom + eps) + x
// ---------------------------------------------------------------------------
__global__ __launch_bounds__(256) void gat_finalize(
    const float* __restrict__ x, const float* __restrict__ denom,
    float* __restrict__ out, int n)
{
  const int i = blockIdx.x * blockDim.x + threadIdx.x;
  const int tot = n * GAT_D;
  if (i >= tot) return;
  const float dv = denom[i >> 7] + GAT_EPS;
  out[i] = out[i] / dv + x[i];
}

// ---------------------------------------------------------------------------
extern "C" void kernel_launch(void* const* d_in, const int* in_sizes, int n_in,
                              void* d_out, int out_size, void* d_ws, size_t ws_size,
                              hipStream_t stream) {
  const float* x   = (const float*)d_in[0];
  const int*   ei  = (const int*)d_in[1];
  const float* w   = (const float*)d_in[2];
  const float* att = (const float*)d_in[3];

  const int N = in_sizes[0] / GAT_D;
  const int E = in_sizes[1] / 2;
  const int* src = ei;
  const int* dst = ei + E;

  float* out = (float*)d_out;

  // Workspace layout
  char* ws = (char*)d_ws;
  float* h = (float*)ws;            ws += (size_t)N * GAT_D * sizeof(float);
  float* a_src = (float*)ws;        ws += (size_t)N * sizeof(float);
  float* a_dst = (float*)ws;        ws += (size_t)N * sizeof(float);
  float* denom = (float*)ws;        ws += (size_t)N * sizeof(float);
  unsigned* gmax = (unsigned*)ws;

  // Zero accumulators (d_out doubles as agg) — in-graph memsets.
  hipMemsetAsync(out, 0, (size_t)N * GAT_D * sizeof(float), stream);
  hipMemsetAsync(denom, 0, (size_t)N * sizeof(float) + sizeof(unsigned), stream);

  // 1) h = x@W (+ fused attention GEMVs)
  {
    dim3 grid((N + 127) / 128);
    gat_gemm_wmma_f32<<<grid, 256, 0, stream>>>(x, w, att, h, a_src, a_dst, N);
  }
  // 2) global logit max
  gat_logit_max<<<1024, 256, 0, stream>>>(src, dst, a_src, a_dst, gmax, E, N);
  // 3) edge softmax-aggregate (wave per edge)
  {
    const int totalWaves = E + N;
    dim3 grid((totalWaves + 7) / 8);
    gat_edge_aggregate<<<grid, 256, 0, stream>>>(src, dst, a_src, a_dst, h,
                                                 gmax, out, denom, E, N);
  }
  // 4) normalize + residual
  {
    const long long tot = (long long)N * GAT_D;
    dim3 grid((unsigned)((tot + 255) / 256));
    gat_finalize<<<grid, 256, 0, stream>>>(x, denom, out, N);
  }
}